// SoftmaxTreeFit_56461640073723
// MI455X (gfx1250) — compile-verified
//
#include <hip/hip_runtime.h>
#include <hip/hip_bf16.h>
#include <stddef.h>

// SoftmaxTreeFit for MI455X (gfx1250, wave32).
// Memory-bound kernel (~430 MB traffic, ~0.2 GFLOP): the einsum is expressed
// as a dense 16x1024x16 GEMM per 16-item tile against a zero-embedded weight
// matrix Wbig[k,l] = (k%16==l) ? W[l,k/16] : 0, executed with
// V_WMMA_F32_16X16X4_F32 (full f32 precision; WMMA throughput is free at
// 0.5 FLOP/byte).

typedef __attribute__((ext_vector_type(2))) float v2f;
typedef __attribute__((ext_vector_type(4))) float v4f;
typedef __attribute__((ext_vector_type(8))) float v8f;

#define KDIM    1024   // V*L
#define LDIM    16
#define VDIM    64
#define WAVES   8      // waves per block, one 16-item tile per wave
#define CHUNK   64     // K-chunk staged through LDS per wave
#define XPITCH  68     // floats; (4r+..)%64 distinct per lane -> conflict-free A reads
#define WPITCH  65     // floats; (l+v)%64 distinct per lane  -> conflict-free W reads

__global__ __launch_bounds__(256)
void softmax_tree_fit_kernel(const float* __restrict__ Gamma,
                             const float* __restrict__ Eta,
                             const float* __restrict__ Wts,
                             const float* __restrict__ X,      // [N,1024]
                             const float* __restrict__ W,      // [16,64]
                             const float* __restrict__ Bias,   // [16]
                             float* __restrict__ out_yhat,     // [N]
                             float* __restrict__ out_w,        // [N]
                             float* __restrict__ out_lamb,     // [N,16]
                             int N, int nTiles)
{
    __shared__ float sW[LDIM * WPITCH];            // padded W copy
    __shared__ float sB[LDIM];
    __shared__ float sX[WAVES][16 * XPITCH];       // per-wave X staging

    const int tid  = threadIdx.x;
    const int lane = tid & 31;
    const int wave = tid >> 5;

    // Stage W (16x64) and bias into LDS with bank-conflict-free padding.
    for (int i = tid; i < LDIM * VDIM; i += 256) {
        const int l = i >> 6, v = i & 63;
        sW[l * WPITCH + v] = W[i];
    }
    if (tid < LDIM) sB[tid] = Bias[tid];

    // weights passthrough (this block's 128 items)
    {
        const int i = blockIdx.x * 128 + tid;
        if (tid < 128 && i < N) out_w[i] = Wts[i];
    }

    __syncthreads();

    const int tile = blockIdx.x * WAVES + wave;
    if (tile >= nTiles) return;                 // wave-uniform: EXEC all-1s inside
    const int base = tile * 16;

    const int col  = lane & 15;                 // N-dim index (= l) for B/C/D
    const int half = lane >> 4;                 // 0: lanes 0-15, 1: lanes 16-31
    const int rx   = half * 2;                  // K sub-offset for A/B operands

    float* Xw = sX[wave];

    v8f acc = {};                               // 16x16 f32 accumulator

    for (int c0 = 0; c0 < KDIM; c0 += CHUNK) {
        // --- stage 16 rows x 64 cols of X into LDS, coalesced b128 loads ---
        #pragma unroll
        for (int i = 0; i < 8; ++i) {
            const int idx = i * 32 + lane;      // 0..255 float4 slots
            const int row = idx >> 4;
            const int q   = idx & 15;
            const v4f val = *(const v4f*)(X + (size_t)(base + row) * KDIM + c0 + q * 4);
            *(v4f*)(Xw + row * XPITCH + q * 4) = val;
        }
        // intra-wave LDS store->load ordering (lanes exchange via LDS)
        asm volatile("s_wait_dscnt 0" ::: "memory");

        const int vbase = c0 >> 4;              // v = k/16 base for this chunk
        #pragma unroll
        for (int t = 0; t < 16; ++t) {
            const int vv   = vbase + (t >> 2);
            const int slot = ((4 * t) & 15) + rx;   // (k % 16) for this lane's B rows

            // A operand: A[M=col][K=4t+rx, 4t+rx+1]  (16x4 f32 layout, 2 VGPRs)
            const v2f a = *(const v2f*)(Xw + col * XPITCH + 4 * t + rx);

            // B operand: Wbig rows 4t+rx, 4t+rx+1 at column l=col
            const float wv = sW[col * WPITCH + vv];     // W[col][vv]
            v2f b;
            b.x = (slot     == col) ? wv : 0.0f;
            b.y = (slot + 1 == col) ? wv : 0.0f;

            acc = __builtin_amdgcn_wmma_f32_16x16x4_f32(
                false, a, false, b, (short)0, acc, false, false);
        }
    }

    // acc layout: component r, lanes 0-15 -> (M=r, l=col); lanes 16-31 -> (M=r+8, l=col)
    const float bval = sB[col];
    float lam[8];
    #pragma unroll
    for (int r = 0; r < 8; ++r) {
        float z = acc[r] + bval;
        // softmax over l (the 16 lanes of this half-wave)
        float m = z;
        m = fmaxf(m, __shfl_xor(m, 1, 32));
        m = fmaxf(m, __shfl_xor(m, 2, 32));
        m = fmaxf(m, __shfl_xor(m, 4, 32));
        m = fmaxf(m, __shfl_xor(m, 8, 32));
        const float e = __expf(z - m);
        float s = e;
        s += __shfl_xor(s, 1, 32);
        s += __shfl_xor(s, 2, 32);
        s += __shfl_xor(s, 4, 32);
        s += __shfl_xor(s, 8, 32);
        lam[r] = e / s;
    }

    #pragma unroll
    for (int r = 0; r < 8; ++r) {
        const int row = base + r + half * 8;
        const size_t off = (size_t)row * LDIM + col;
        const float lv = lam[r];
        out_lamb[off] = lv;                       // coalesced 64B per half-wave

        float g = Gamma[off] * lv;
        float e = Eta[off]   * lv;
        g += __shfl_xor(g, 1, 32);
        g += __shfl_xor(g, 2, 32);
        g += __shfl_xor(g, 4, 32);
        g += __shfl_xor(g, 8, 32);
        e += __shfl_xor(e, 1, 32);
        e += __shfl_xor(e, 2, 32);
        e += __shfl_xor(e, 4, 32);
        e += __shfl_xor(e, 8, 32);
        if (col == r) out_yhat[row] = g / e;      // lane r -> row base+r, lane 16+r -> base+8+r
    }
}

extern "C" void kernel_launch(void* const* d_in, const int* in_sizes, int n_in,
                              void* d_out, int out_size, void* d_ws, size_t ws_size,
                              hipStream_t stream) {
    // setup_inputs order: y_item, Gamma_item, eta_item, weights_item,
    //                     softmax_structure_item, W, b   (all float32)
    const float* Gamma = (const float*)d_in[1];
    const float* Eta   = (const float*)d_in[2];
    const float* Wts   = (const float*)d_in[3];
    const float* X     = (const float*)d_in[4];
    const float* W     = (const float*)d_in[5];
    const float* Bias  = (const float*)d_in[6];

    const int N = in_sizes[0];            // B*N = 100000 (divisible by 16)
    float* out      = (float*)d_out;
    float* out_yhat = out;                // [N]
    float* out_w    = out + N;            // [N]
    float* out_lamb = out + 2 * (size_t)N;// [N,16]

    const int nTiles = N / 16;
    const int blocks = (nTiles + WAVES - 1) / WAVES;
    hipLaunchKernelGGL(softmax_tree_fit_kernel, dim3(blocks), dim3(256), 0, stream,
                       Gamma, Eta, Wts, X, W, Bias,
                       out_yhat, out_w, out_lamb, N, nTiles);
}